// IndexBEVProjector_481036337894
// MI455X (gfx1250) — compile-verified
//
#include <hip/hip_runtime.h>

typedef __attribute__((ext_vector_type(2))) float v2f;
typedef __attribute__((ext_vector_type(4))) float f4;
typedef __attribute__((ext_vector_type(8))) float v8f;

#define KPTS 16384   // 128*128 BEV points
#define NCAM 6
#define IMH 32
#define IMW 88
#define NCH 128
#define NGP 9

// ---------------------------------------------------------------------------
// Kernel 1: M = I4 @ E  (6 tiny 4x4 matmuls)
// ---------------------------------------------------------------------------
__global__ void compute_M(const float* __restrict__ I, const float* __restrict__ E,
                          float* __restrict__ Mw) {
    int t = threadIdx.x;
    if (t >= NCAM * 16) return;
    int n = t >> 4;
    int r = t & 15;
    int i = r >> 2;
    int j = r & 3;
    float val;
    if (i == 3) {
        val = E[n * 16 + 12 + j];                 // I4 row3 = [0,0,0,1]
    } else {
        val = 0.0f;
        for (int l = 0; l < 3; ++l)               // I4[i][3] == 0
            val += I[n * 9 + i * 3 + l] * E[n * 16 + l * 4 + j];
    }
    Mw[n * 16 + i * 4 + j] = val;
}

// ---------------------------------------------------------------------------
// Kernel 2: sp = einsum('ij,kj->ki', M, pts_h) via V_WMMA_F32_16X16X4_F32.
// One wave transforms 16 points: A = pts_h (16x4), B cols 0..3 = M^T (4x16).
// Then per-point projective divide, mask, pixel rounding; emit packed px/py
// and the mask floats (mask goes straight into d_out tail).
// ---------------------------------------------------------------------------
__global__ __launch_bounds__(32) void bev_transform_wmma(
    const float* __restrict__ bev,   // (3,128,128): channel-major
    const float* __restrict__ Mmat,  // (6,4,4)
    int* __restrict__ packed,        // (6,16384) packed px|py<<16
    float* __restrict__ maskout)     // (6,16384) 0/1 floats
{
    int group = blockIdx.x;                   // 0 .. 6*16384/16 - 1
    int n  = group / (KPTS / 16);
    int k0 = (group % (KPTS / 16)) * 16;
    int lane = threadIdx.x;
    int p  = lane & 15;
    int hi = lane >> 4;
    int kk = k0 + p;

    // A-matrix 16x4 f32 layout: lanes 0-15 -> K=0 (v0), K=1 (v1);
    //                           lanes 16-31 -> K=2 (v0), K=3 (v1).
    // pts_h = (x, y, -1, 1)
    v2f a;
    a.x = hi ? -1.0f : bev[kk];               // K=0 or K=2
    a.y = hi ?  1.0f : bev[KPTS + kk];        // K=1 or K=3

    // B-matrix 4x16: row striped across lanes within a VGPR, halves split like A.
    // B[j][i] = M[i][j]; columns >=4 are zero (unused output columns).
    const float* M = Mmat + n * 16;
    v2f b;
    if (p < 4) {
        b.x = M[p * 4 + (hi ? 2 : 0)];
        b.y = M[p * 4 + (hi ? 3 : 1)];
    } else {
        b.x = 0.0f;
        b.y = 0.0f;
    }

    v8f c = {};
    v8f d = __builtin_amdgcn_wmma_f32_16x16x4_f32(
        /*neg_a=*/false, a, /*neg_b=*/false, b,
        /*c_mod=*/(short)0, c, /*reuse_a=*/false, /*reuse_b=*/false);

    // D layout (16x16 f32): VGPR r, lanes 0-15 -> row r, lanes 16-31 -> row r+8.
    __shared__ float sd[8 * 32];
#pragma unroll
    for (int r = 0; r < 8; ++r) sd[r * 32 + lane] = d[r];
    __syncthreads();

    if (lane < 16) {
        int r    = lane & 7;
        int half = lane >> 3;
        float sx = sd[r * 32 + half * 16 + 0];
        float sy = sd[r * 32 + half * 16 + 1];
        float sz = sd[r * 32 + half * 16 + 2];

        bool m   = sz > 1e-5f;
        float zc = fmaxf(sz, 1e-5f);
        float u  = sx / zc / 704.0f;          // / Wimg
        float v  = sy / zc / 256.0f;          // / Himg
        m = m && (u > 0.0f) && (u < 1.0f) && (v > 0.0f) && (v < 1.0f);

        int px = __float2int_rn(u * (float)IMW);   // round-half-even == jnp.round
        int py = __float2int_rn(v * (float)IMH);
        // clamp-safe range: final index is clip(px+dx,0,86..87) with dx in {-1,0,1}
        px = min(max(px, -3), IMW + 2);
        py = min(max(py, -3), IMH + 2);

        int idx = n * KPTS + k0 + lane;
        packed[idx]  = (px & 0xffff) | (py << 16);
        maskout[idx] = m ? 1.0f : 0.0f;
    }
}

// ---------------------------------------------------------------------------
// Kernel 3: NCHW -> NHWC transpose of images (8.6 MB, L2-resident afterwards)
// ---------------------------------------------------------------------------
__global__ __launch_bounds__(256) void transpose_kernel(
    const float* __restrict__ images, float* __restrict__ imt) {
    int idx = blockIdx.x * 256 + threadIdx.x;
    const int total = NCAM * NCH * IMH * IMW;
    if (idx >= total) return;
    int ch   = idx & (NCH - 1);
    int rest = idx >> 7;
    int x = rest % IMW;  rest /= IMW;
    int y = rest % IMH;
    int n = rest / IMH;
    imt[idx] = images[((n * NCH + ch) * IMH + y) * IMW + x];
}

// ---------------------------------------------------------------------------
// Kernel 4: gather. One wave per (n,k,g) vector: 32 lanes x float4 = 512 B.
// Loads hit L2 (images only 8.6 MB); output streamed with NT stores.
// ---------------------------------------------------------------------------
__global__ __launch_bounds__(256) void gather_kernel(
    const f4* __restrict__ imt,       // (6,32,88,128) as f4, 32 f4 per pixel
    const int* __restrict__ packed,
    f4* __restrict__ out)             // feats as f4
{
    long long gid = (long long)blockIdx.x * 256 + threadIdx.x;
    int part = (int)(gid & 31);               // f4 slot within 128-ch vector
    long long v = gid >> 5;                   // vector id = (n*KPTS + k)*9 + g
    if (v >= (long long)NCAM * KPTS * NGP) return;

    int g        = (int)(v % NGP);
    long long nk = v / NGP;
    int k = (int)(nk % KPTS);
    int n = (int)(nk / KPTS);

    int pk = packed[n * KPTS + k];
    int px = (int)(short)(pk & 0xffff);
    int py = pk >> 16;
    int dx = g % 3 - 1;
    int dy = g / 3 - 1;
    int xs = min(max(px + dx, 0), IMW - 1);
    int ys = min(max(py + dy, 0), IMH - 1);

    const f4* src = imt + (long long)((n * IMH + ys) * IMW + xs) * 32 + part;
    f4 val = *src;
    __builtin_nontemporal_store(val, out + v * 32 + part);
}

// ---------------------------------------------------------------------------
extern "C" void kernel_launch(void* const* d_in, const int* in_sizes, int n_in,
                              void* d_out, int out_size, void* d_ws, size_t ws_size,
                              hipStream_t stream) {
    const float* bev    = (const float*)d_in[0];   // (3,128,128)
    const float* images = (const float*)d_in[1];   // (6,128,32,88)
    const float* I      = (const float*)d_in[2];   // (1,6,3,3)
    const float* E      = (const float*)d_in[3];   // (1,6,4,4)
    float* out = (float*)d_out;

    // workspace layout (floats): [0..127] M, [128..] packed px/py, then NHWC images
    float* wsf    = (float*)d_ws;
    float* Mw     = wsf;
    int*   packed = (int*)(wsf + 128);
    float* imt    = wsf + 128 + NCAM * KPTS;

    const long long FEATS = (long long)NCAM * KPTS * NGP * NCH;  // 113246208
    float* maskout = out + FEATS;

    hipLaunchKernelGGL(compute_M, dim3(1), dim3(96), 0, stream, I, E, Mw);
    hipLaunchKernelGGL(bev_transform_wmma, dim3(NCAM * KPTS / 16), dim3(32), 0, stream,
                       bev, Mw, packed, maskout);
    hipLaunchKernelGGL(transpose_kernel, dim3((NCAM * NCH * IMH * IMW) / 256), dim3(256),
                       0, stream, images, imt);
    hipLaunchKernelGGL(gather_kernel, dim3((unsigned)((long long)NCAM * KPTS * NGP * 32 / 256)),
                       dim3(256), 0, stream, (const f4*)imt, packed, (f4*)out);
}